// VAE_Delta_19224273616901
// MI455X (gfx1250) — compile-verified
//
#include <hip/hip_runtime.h>
#include <hip/hip_bf16.h>
#include <cmath>

// MI455X / gfx1250, wave32. FP32 WMMA (V_WMMA_F32_16X16X4_F32) keeps full
// reference precision; workload is HBM-bound (~220MB @ 23.3TB/s ~ 9.5us),
// so fusing the per-bead chain in LDS matters more than matrix-core rate.

typedef float v2f __attribute__((ext_vector_type(2)));
typedef float v8f __attribute__((ext_vector_type(8)));

#define WMMA_F32X4(a, b, c) \
  __builtin_amdgcn_wmma_f32_16x16x4_f32(false, (a), false, (b), (short)0, (c), false, false)

#define NBEADS 32768
#define FC 64
#define NMAX 8
#define GB 16   // beads per workgroup tile (= WMMA N dim)

// ---------------------------------------------------------------------------
// Kernel 1: per-bead VN MLP + VN-ReLU + VN-LayerNorm + tiny attention.
// One 256-thread (8-wave) block processes GB=16 beads per iteration.
// vn_linear over 'i' is a GEMM with columns = beads, one plane per d=0..2.
// ---------------------------------------------------------------------------
__global__ __launch_bounds__(256) void vn_bead_kernel(
    const float* __restrict__ Z_V, const float* __restrict__ Q,
    const float* __restrict__ W1,  const float* __restrict__ W2,
    const float* __restrict__ Wd,  const float* __restrict__ g1,
    const float* __restrict__ b1,  const float* __restrict__ Wq,
    const float* __restrict__ bq,  const float* __restrict__ Wk,
    const float* __restrict__ bk,  const float* __restrict__ Wv,
    const float* __restrict__ bv,  const float* __restrict__ Wo,
    const float* __restrict__ bo,  float* __restrict__ attn_out)
{
  // 15360 floats = 60KB static LDS, buffers aliased across pipeline stages.
  __shared__ float smem[15360];
  float* sT1 = smem;           // [3][128][16] = 6144   (later: K | V, 3072 each)
  float* sT2 = smem + 6144;    // [3][64][16]  = 3072   (post-relu result in place)
  float* sD  = smem + 9216;    // [3][64][16]  = 3072   (later: Z)
  float* sX  = smem + 12288;   // [3][64][16]  = 3072   (later: reduction scratch)
  float* sK  = sT1;
  float* sV  = sT1 + 3072;
  float* sZ  = sD;
  float* sRed  = sX;           // 256
  float* sRed2 = sX + 256;     // 256

  const int tid  = threadIdx.x;
  const int wave = tid >> 5;
  const int lane = tid & 31;
  const int hrow = lane & 15;           // tile row (A) / tile col (B,C,D)
  const int koff = (lane >> 4) << 1;    // K sub-offset: 0 (lanes 0-15) / 2 (16-31)
  const int rbase = (lane >> 4) << 3;   // C/D row base: 0 / 8

  for (int g = blockIdx.x; g < NBEADS / GB; g += gridDim.x) {
    const int bead0 = g * GB;

    // ---- Stage A: load Z_V tile -> sX[d][c][b] (coalesced read, LDS scatter)
    for (int i = tid; i < 3 * FC * GB; i += 256) {
      int bl = i / 192, rem = i % 192;
      int c = rem / 3, d = rem % 3;
      sX[d * 1024 + c * 16 + bl] = Z_V[(size_t)bead0 * 192 + i];
    }
    // Prefetch next group's Z_V (global_prefetch_b8).
    {
      int nb0 = (g + gridDim.x) * GB;
      if (nb0 < NBEADS)
        __builtin_prefetch(Z_V + (size_t)nb0 * 192 + tid * 12, 0, 3);
    }
    __syncthreads();

    // ---- Stage B: T1 = W1 @ X  (M=128 -> 8 row tiles, K=64, N=16), x3 planes.
    // Wave w owns row tile w for all d. Weights stream from global (L2-hot).
    for (int d = 0; d < 3; ++d) {
      const int row0 = wave * 16;
      v8f acc = {};
#pragma unroll 4
      for (int ks = 0; ks < 16; ++ks) {
        int k = ks * 4 + koff;
        v2f a, b;
        a.x = W1[(row0 + hrow) * 64 + k];
        a.y = W1[(row0 + hrow) * 64 + k + 1];
        b.x = sX[d * 1024 + k * 16 + hrow];
        b.y = sX[d * 1024 + (k + 1) * 16 + hrow];
        acc = WMMA_F32X4(a, b, acc);
      }
#pragma unroll
      for (int r = 0; r < 8; ++r)
        sT1[d * 2048 + (row0 + rbase + r) * 16 + hrow] = acc[r];
    }
    __syncthreads();

    // ---- Stage C: T2 = W2 @ T1  (M=64 -> 4 tiles, K=128), x3 planes = 12 jobs.
    for (int job = wave; job < 12; job += 8) {
      const int t = job & 3, d = job >> 2;
      const int row0 = t * 16;
      v8f acc = {};
#pragma unroll 4
      for (int ks = 0; ks < 32; ++ks) {
        int k = ks * 4 + koff;
        v2f a, b;
        a.x = W2[(row0 + hrow) * 128 + k];
        a.y = W2[(row0 + hrow) * 128 + k + 1];
        b.x = sT1[d * 2048 + k * 16 + hrow];
        b.y = sT1[d * 2048 + (k + 1) * 16 + hrow];
        acc = WMMA_F32X4(a, b, acc);
      }
#pragma unroll
      for (int r = 0; r < 8; ++r)
        sT2[d * 1024 + (row0 + rbase + r) * 16 + hrow] = acc[r];
    }
    __syncthreads();

    // ---- Stage D: Dm = Wd @ T2  (M=64, K=64) x3 = 12 jobs.
    for (int job = wave; job < 12; job += 8) {
      const int t = job & 3, d = job >> 2;
      const int row0 = t * 16;
      v8f acc = {};
#pragma unroll 4
      for (int ks = 0; ks < 16; ++ks) {
        int k = ks * 4 + koff;
        v2f a, b;
        a.x = Wd[(row0 + hrow) * 64 + k];
        a.y = Wd[(row0 + hrow) * 64 + k + 1];
        b.x = sT2[d * 1024 + k * 16 + hrow];
        b.y = sT2[d * 1024 + (k + 1) * 16 + hrow];
        acc = WMMA_F32X4(a, b, acc);
      }
#pragma unroll
      for (int r = 0; r < 8; ++r)
        sD[d * 1024 + (row0 + rbase + r) * 16 + hrow] = acc[r];
    }
    __syncthreads();

    // ---- Stage E: VN-ReLU + residual (elementwise over 64x16, write in place).
    for (int i = tid; i < 1024; i += 256) {
      float x0 = sT2[i], x1 = sT2[1024 + i], x2 = sT2[2048 + i];
      float d0 = sD[i],  d1 = sD[1024 + i],  d2 = sD[2048 + i];
      float dot = x0 * d0 + x1 * d1 + x2 * d2;
      float dn  = d0 * d0 + d1 * d1 + d2 * d2;
      float s = (dot >= 0.f) ? 0.f : dot / (dn + 1e-6f);
      sT2[i]        = x0 - s * d0 + sX[i];
      sT2[1024 + i] = x1 - s * d1 + sX[1024 + i];
      sT2[2048 + i] = x2 - s * d2 + sX[2048 + i];
    }
    __syncthreads();

    // ---- Stage F: VN-LayerNorm per bead over 64 channels -> sZ.
    {
      const int b = tid & 15, sub = tid >> 4;   // 16 threads per bead, 4 chans each
      float ps = 0.f, ps2 = 0.f;
#pragma unroll
      for (int j = 0; j < 4; ++j) {
        int idx = (sub * 4 + j) * 16 + b;
        float x0 = sT2[idx], x1 = sT2[1024 + idx], x2 = sT2[2048 + idx];
        float n = sqrtf(x0 * x0 + x1 * x1 + x2 * x2 + 1e-12f);
        ps += n; ps2 += n * n;
      }
      sRed[sub * 16 + b] = ps;
      sRed2[sub * 16 + b] = ps2;
      __syncthreads();
      float sum = 0.f, sum2 = 0.f;
#pragma unroll
      for (int s2 = 0; s2 < 16; ++s2) { sum += sRed[s2 * 16 + b]; sum2 += sRed2[s2 * 16 + b]; }
      float mu  = sum  * (1.f / 64.f);
      float var = sum2 * (1.f / 64.f) - mu * mu;
      float rs  = rsqrtf(var + 1e-5f);
#pragma unroll
      for (int j = 0; j < 4; ++j) {
        int c = sub * 4 + j, idx = c * 16 + b;
        float x0 = sT2[idx], x1 = sT2[1024 + idx], x2 = sT2[2048 + idx];
        float n = sqrtf(x0 * x0 + x1 * x1 + x2 * x2 + 1e-12f);
        float ln = (n - mu) * rs * g1[c] + b1[c];
        float sc = ln / (n + 1e-5f);                 // vn_layernorm eps = 1e-5
        sZ[idx] = x0 * sc; sZ[1024 + idx] = x1 * sc; sZ[2048 + idx] = x2 * sc;
      }
    }
    __syncthreads();

    // ---- Stage G: k = Z @ Wk.T + bk, v = Z @ Wv.T + bv  -> sK, sV.
    for (int i = tid; i < 1024; i += 256) {
      float z0 = sZ[i], z1 = sZ[1024 + i], z2 = sZ[2048 + i];
#pragma unroll
      for (int e = 0; e < 3; ++e) {
        sK[e * 1024 + i] = Wk[e * 3] * z0 + Wk[e * 3 + 1] * z1 + Wk[e * 3 + 2] * z2 + bk[e];
        sV[e * 1024 + i] = Wv[e * 3] * z0 + Wv[e * 3 + 1] * z1 + Wv[e * 3 + 2] * z2 + bv[e];
      }
    }
    __syncthreads();

    // ---- Stage H: attention, one thread per (bead, query-row). Invalid rows
    // (slot >= group_size) are never gathered by flat_idx, so no masking needed.
    if (tid < GB * NMAX) {
      const int b = tid >> 3, qi = tid & 7;
      const float* qp = Q + ((size_t)(bead0 + b) * NMAX + qi) * 3;
      float q0 = Wq[0] * qp[0] + Wq[1] * qp[1] + Wq[2] * qp[2] + bq[0];
      float q1 = Wq[3] * qp[0] + Wq[4] * qp[1] + Wq[5] * qp[2] + bq[1];
      float q2 = Wq[6] * qp[0] + Wq[7] * qp[1] + Wq[8] * qp[2] + bq[2];
      const float inv3 = 0.5773502691896258f;  // 1/sqrt(3)
      float mx = -1e30f;
      for (int c = 0; c < FC; ++c) {
        int idx = c * 16 + b;
        float s = (q0 * sK[idx] + q1 * sK[1024 + idx] + q2 * sK[2048 + idx]) * inv3;
        mx = fmaxf(mx, s);
      }
      float den = 0.f, a0 = 0.f, a1 = 0.f, a2 = 0.f;
      for (int c = 0; c < FC; ++c) {
        int idx = c * 16 + b;
        float s = (q0 * sK[idx] + q1 * sK[1024 + idx] + q2 * sK[2048 + idx]) * inv3;
        float e = expf(s - mx);
        den += e;
        a0 += e * sV[idx]; a1 += e * sV[1024 + idx]; a2 += e * sV[2048 + idx];
      }
      float r = 1.f / den;
      float o0 = a0 * r, o1 = a1 * r, o2 = a2 * r;
      float* op = attn_out + ((size_t)(bead0 + b) * NMAX + qi) * 3;
      op[0] = Wo[0] * o0 + Wo[1] * o1 + Wo[2] * o2 + bo[0];
      op[1] = Wo[3] * o0 + Wo[4] * o1 + Wo[5] * o2 + bo[1];
      op[2] = Wo[6] * o0 + Wo[7] * o1 + Wo[8] * o2 + bo[2];
    }
    __syncthreads();
  }
}

// ---------------------------------------------------------------------------
// Kernel 2: gather via flat_idx, per-point 3->64->3 VN-FF + VN-LayerNorm -> x_cc
// ---------------------------------------------------------------------------
__global__ __launch_bounds__(256) void point_ff_kernel(
    const float* __restrict__ attn_out, const int* __restrict__ flat_idx,
    const float* __restrict__ Wa, const float* __restrict__ Wb,
    const float* __restrict__ Wc, const float* __restrict__ g2,
    const float* __restrict__ b2, float* __restrict__ x_cc, int M)
{
  int m = blockIdx.x * 256 + threadIdx.x;
  if (m >= M) return;
  int fi = flat_idx[m];
  float x0 = attn_out[(size_t)fi * 3 + 0];
  float x1 = attn_out[(size_t)fi * 3 + 1];
  float x2 = attn_out[(size_t)fi * 3 + 2];
  // y = Wb @ (Wa @ x): fuse the 64-dim hidden.
  float y0 = 0.f, y1 = 0.f, y2 = 0.f;
#pragma unroll 8
  for (int c = 0; c < FC; ++c) {
    float t = Wa[c * 3] * x0 + Wa[c * 3 + 1] * x1 + Wa[c * 3 + 2] * x2;
    y0 += Wb[c] * t; y1 += Wb[64 + c] * t; y2 += Wb[128 + c] * t;
  }
  // vn_relu(y, Wc)
  float d0 = Wc[0] * y0 + Wc[1] * y1 + Wc[2] * y2;
  float d1 = Wc[3] * y0 + Wc[4] * y1 + Wc[5] * y2;
  float d2 = Wc[6] * y0 + Wc[7] * y1 + Wc[8] * y2;
  float dot = y0 * d0 + y1 * d1 + y2 * d2;
  float dn  = d0 * d0 + d1 * d1 + d2 * d2;
  float s = (dot >= 0.f) ? 0.f : dot / (dn + 1e-6f);
  x0 += y0 - s * d0; x1 += y1 - s * d1; x2 += y2 - s * d2;
  // vn_layernorm on (M,3,1): per-component norm collapses to |x|.
  float n0 = sqrtf(x0 * x0 + 1e-12f);
  float n1 = sqrtf(x1 * x1 + 1e-12f);
  float n2 = sqrtf(x2 * x2 + 1e-12f);
  float mu  = (n0 + n1 + n2) * (1.f / 3.f);
  float var = (n0 * n0 + n1 * n1 + n2 * n2) * (1.f / 3.f) - mu * mu;
  float rs  = rsqrtf(var + 1e-5f);
  x_cc[(size_t)m * 3 + 0] = x0 * (((n0 - mu) * rs * g2[0] + b2[0]) / (n0 + 1e-5f));
  x_cc[(size_t)m * 3 + 1] = x1 * (((n1 - mu) * rs * g2[1] + b2[1]) / (n1 + 1e-5f));
  x_cc[(size_t)m * 3 + 2] = x2 * (((n2 - mu) * rs * g2[2] + b2[2]) / (n2 + 1e-5f));
}

// ---------------------------------------------------------------------------
// Kernel 3: h_cc = ref_feat(Mx64) @ W_emb.T(64x256) + b_emb, FP32 WMMA.
// Block = 8 waves covering a 32x256 output tile; each wave: 16 rows x 64 cols,
// A fragments (16xK=64) resident in 32 VGPRs, bias folded into accumulator.
// ---------------------------------------------------------------------------
__global__ __launch_bounds__(256) void hcc_gemm_kernel(
    const float* __restrict__ A,     // ref_feat, M x 64
    const float* __restrict__ W,     // W_emb, 256 x 64
    const float* __restrict__ bias,  // b_emb, 256
    float* __restrict__ H, int M)    // M x 256
{
  const int tid  = threadIdx.x;
  const int wave = tid >> 5;
  const int lane = tid & 31;
  const int hrow = lane & 15;
  const int koff = (lane >> 4) << 1;
  const int rbase = (lane >> 4) << 3;

  const int mbase = blockIdx.x * 32 + (wave & 1) * 16;
  const int cbase = (wave >> 1) * 64;
  int mrow = mbase + hrow;
  int mclamp = (mrow < M) ? mrow : (M - 1);

  v2f afr[16];
#pragma unroll
  for (int ks = 0; ks < 16; ++ks) {
    int k = ks * 4 + koff;
    afr[ks].x = A[(size_t)mclamp * 64 + k];
    afr[ks].y = A[(size_t)mclamp * 64 + k + 1];
  }

#pragma unroll
  for (int ct = 0; ct < 4; ++ct) {
    const int col = cbase + ct * 16 + hrow;
    v8f acc;
    float bs = bias[col];
#pragma unroll
    for (int r = 0; r < 8; ++r) acc[r] = bs;
#pragma unroll
    for (int ks = 0; ks < 16; ++ks) {
      int k = ks * 4 + koff;
      v2f b;
      b.x = W[(size_t)col * 64 + k];      // B[k][n] = W_emb[n][k]
      b.y = W[(size_t)col * 64 + k + 1];
      acc = WMMA_F32X4(afr[ks], b, acc);
    }
#pragma unroll
    for (int r = 0; r < 8; ++r) {
      int mm = mbase + rbase + r;
      if (mm < M) H[(size_t)mm * 256 + col] = acc[r];
    }
  }
}

// ---------------------------------------------------------------------------
extern "C" void kernel_launch(void* const* d_in, const int* in_sizes, int n_in,
                              void* d_out, int out_size, void* d_ws, size_t ws_size,
                              hipStream_t stream) {
  const float* Z_V   = (const float*)d_in[0];
  const float* Q     = (const float*)d_in[1];
  const float* rfeat = (const float*)d_in[2];
  const float* W1    = (const float*)d_in[3];
  const float* W2    = (const float*)d_in[4];
  const float* Wd    = (const float*)d_in[5];
  const float* g1    = (const float*)d_in[6];
  const float* b1    = (const float*)d_in[7];
  const float* Wq    = (const float*)d_in[8];
  const float* bq    = (const float*)d_in[9];
  const float* Wk    = (const float*)d_in[10];
  const float* bk    = (const float*)d_in[11];
  const float* Wv    = (const float*)d_in[12];
  const float* bv    = (const float*)d_in[13];
  const float* Wo    = (const float*)d_in[14];
  const float* bo    = (const float*)d_in[15];
  const float* Wa    = (const float*)d_in[16];
  const float* Wb    = (const float*)d_in[17];
  const float* Wc    = (const float*)d_in[18];
  const float* g2    = (const float*)d_in[19];
  const float* b2    = (const float*)d_in[20];
  const float* W_emb = (const float*)d_in[21];
  const float* b_emb = (const float*)d_in[22];
  const int*   fidx  = (const int*)d_in[24];

  const int M = in_sizes[24];             // flat_idx count
  float* out   = (float*)d_out;
  float* x_cc  = out;                     // M x 3
  float* h_cc  = out + (size_t)M * 3;     // M x 256
  float* attnw = (float*)d_ws;            // N x 8 x 3 floats = 3 MB scratch

  vn_bead_kernel<<<512, 256, 0, stream>>>(Z_V, Q, W1, W2, Wd, g1, b1,
                                          Wq, bq, Wk, bk, Wv, bv, Wo, bo, attnw);
  point_ff_kernel<<<(M + 255) / 256, 256, 0, stream>>>(attnw, fidx, Wa, Wb, Wc,
                                                       g2, b2, x_cc, M);
  hcc_gemm_kernel<<<(M + 31) / 32, 256, 0, stream>>>(rfeat, W_emb, b_emb, h_cc, M);
}